// homo_meta_path_layer_min_2680059592916
// MI455X (gfx1250) — compile-verified
//
#include <hip/hip_runtime.h>
#include <hip/hip_bf16.h>
#include <math.h>

#define NND   4096        // N nodes
#define DIM   256         // feature dim D
#define HID   256         // hidden H
#define TAUF  0.8f
#define LAMF  0.5f
#define TPB   256

// GEMM tiling
#define BM    128         // block tile M (8 waves x 16)
#define BN    128         // block tile N
#define KC    32          // K slice staged in LDS
#define RS_A  36          // LDS row stride (words) for 32-float rows (+pad, 16B aligned)
#define RS_BN 132         // LDS row stride (words) for 128-float rows (+pad, 16B aligned)

typedef __attribute__((ext_vector_type(2))) float v2f;
typedef __attribute__((ext_vector_type(8))) float v8f;

enum { ACT_NONE = 0, ACT_TANH = 1, ACT_RELU = 2, ACT_ELU = 3 };

__device__ __forceinline__ float apply_act(float v, int act) {
  switch (act) {
    case ACT_TANH: return tanhf(v);
    case ACT_RELU: return v > 0.f ? v : 0.f;
    case ACT_ELU:  return v > 0.f ? v : (expf(v) - 1.f);
    default:       return v;
  }
}

// ---- CDNA5 async global->LDS copy (ASYNCcnt) -------------------------------
__device__ __forceinline__ void async_load16(const float* gptr, unsigned lds_off) {
  unsigned long long ga = (unsigned long long)(uintptr_t)gptr;
  asm volatile("global_load_async_to_lds_b128 %0, %1, off"
               :: "v"(lds_off), "v"(ga) : "memory");
}
__device__ __forceinline__ unsigned lds_off(const void* p) {
  // generic LDS pointer: addr[31:0] is the LDS byte offset
  return (unsigned)(uintptr_t)p;
}
__device__ __forceinline__ void wait_async8() {
  asm volatile("s_wait_asynccnt 0x8" ::: "memory");
}
__device__ __forceinline__ void wait_async0() {
  asm volatile("s_wait_asynccnt 0x0" ::: "memory");
}

__device__ __forceinline__ float block_reduce_sum(float v, float* sm) {
  int t = threadIdx.x;
  sm[t] = v;
  __syncthreads();
  for (int s = TPB >> 1; s > 0; s >>= 1) {
    if (t < s) sm[t] += sm[t + s];
    __syncthreads();
  }
  float r = sm[0];
  __syncthreads();
  return r;
}

// ---------------------------------------------------------------------------
// C[M,Nc] = act( alpha * A[M,K] @ W[Nc,K]^T + bias[col] (+ C) )
// 256 threads = 8 waves; block tile 128x128; each wave computes 16x128.
// Pipeline: global -> (async b128, double buffered) -> LDS -> ds_load -> WMMA.
// ---------------------------------------------------------------------------
__global__ void __launch_bounds__(256) gemm_nt_f32(
    const float* __restrict__ A, int lda,
    const float* __restrict__ W, int ldw,
    float* C, int ldc,
    const float* __restrict__ bias,
    float alpha, int accumulate, int act, int K) {
  __shared__ float smA[2][BM * RS_A];
  __shared__ float smB[2][BN * RS_A];

  const int tid  = threadIdx.x;
  const int wave = tid >> 5;
  const int lane = tid & 31;
  const int half = lane >> 4;
  const int l    = lane & 15;
  const int mrow0 = blockIdx.y * BM;
  const int nrow0 = blockIdx.x * BN;

  v8f acc[8];
#pragma unroll
  for (int t = 0; t < 8; ++t)
#pragma unroll
    for (int r = 0; r < 8; ++r) acc[t][r] = 0.f;

  // stage issuer: 128 rows x 32 floats per matrix, 16B chunks, 8 ops/thread
  auto issue = [&](int buf, int kc) {
#pragma unroll
    for (int i = 0; i < 4; ++i) {
      const int c   = tid + TPB * i;   // 0..1023
      const int row = c >> 3;
      const int kch = (c & 7) << 2;
      async_load16(A + (size_t)(mrow0 + row) * lda + kc + kch,
                   lds_off(&smA[buf][row * RS_A + kch]));
      async_load16(W + (size_t)(nrow0 + row) * ldw + kc + kch,
                   lds_off(&smB[buf][row * RS_A + kch]));
    }
  };

  const int nstages = K / KC;
  issue(0, 0);
  for (int s = 0; s < nstages; ++s) {
    const int cur = s & 1;
    if (s + 1 < nstages) {
      issue(cur ^ 1, (s + 1) * KC);
      wait_async8();            // current buffer's 8 ops complete (in-order)
    } else {
      wait_async0();
    }
    __syncthreads();

    const float* As = smA[cur];
    const float* Bs = smB[cur];
    const int arow = (wave * 16 + l) * RS_A;
#pragma unroll
    for (int k = 0; k < KC; k += 4) {
      const v2f a = *(const v2f*)&As[arow + k + 2 * half];
#pragma unroll
      for (int t = 0; t < 8; ++t) {
        const v2f b = *(const v2f*)&Bs[(16 * t + l) * RS_A + k + 2 * half];
        acc[t] = __builtin_amdgcn_wmma_f32_16x16x4_f32(
            false, a, false, b, (short)0, acc[t], false, false);
      }
    }
    __syncthreads();            // done reading before next stage overwrites
  }

#pragma unroll
  for (int t = 0; t < 8; ++t) {
    const int col = nrow0 + 16 * t + l;
    const float bv = bias ? bias[col] : 0.f;
#pragma unroll
    for (int r = 0; r < 8; ++r) {
      const int row = mrow0 + wave * 16 + r + 8 * half;
      const size_t idx = (size_t)row * ldc + col;
      float v = alpha * acc[t][r] + bv;
      if (accumulate) v += C[idx];
      C[idx] = apply_act(v, act);
    }
  }
}

// ---------------------------------------------------------------------------
// C[M,Nc] = A[M,K] @ B[K,Nc]  (row-major B; used for  P @ feats)
// ---------------------------------------------------------------------------
__global__ void __launch_bounds__(256) gemm_nn_f32(
    const float* __restrict__ A, int lda,
    const float* __restrict__ B, int ldb,
    float* C, int ldc, int K) {
  __shared__ float smA[2][BM * RS_A];          // 128 rows x 32 k-floats
  __shared__ float smB[2][KC * RS_BN];         // 32 k-rows x 128 n-floats

  const int tid  = threadIdx.x;
  const int wave = tid >> 5;
  const int lane = tid & 31;
  const int half = lane >> 4;
  const int l    = lane & 15;
  const int mrow0 = blockIdx.y * BM;
  const int ncol0 = blockIdx.x * BN;

  v8f acc[8];
#pragma unroll
  for (int t = 0; t < 8; ++t)
#pragma unroll
    for (int r = 0; r < 8; ++r) acc[t][r] = 0.f;

  auto issue = [&](int buf, int kc) {
#pragma unroll
    for (int i = 0; i < 4; ++i) {
      const int c = tid + TPB * i;             // 0..1023
      const int arow = c >> 3;
      const int akch = (c & 7) << 2;
      async_load16(A + (size_t)(mrow0 + arow) * lda + kc + akch,
                   lds_off(&smA[buf][arow * RS_A + akch]));
      const int brow = c >> 5;                 // 0..31
      const int bnch = (c & 31) << 2;          // 0..124
      async_load16(B + (size_t)(kc + brow) * ldb + ncol0 + bnch,
                   lds_off(&smB[buf][brow * RS_BN + bnch]));
    }
  };

  const int nstages = K / KC;
  issue(0, 0);
  for (int s = 0; s < nstages; ++s) {
    const int cur = s & 1;
    if (s + 1 < nstages) {
      issue(cur ^ 1, (s + 1) * KC);
      wait_async8();
    } else {
      wait_async0();
    }
    __syncthreads();

    const float* As = smA[cur];
    const float* Bs = smB[cur];
    const int arow = (wave * 16 + l) * RS_A;
#pragma unroll
    for (int k = 0; k < KC; k += 4) {
      const v2f a = *(const v2f*)&As[arow + k + 2 * half];
#pragma unroll
      for (int t = 0; t < 8; ++t) {
        v2f b;
        b.x = Bs[(k + 2 * half + 0) * RS_BN + 16 * t + l];
        b.y = Bs[(k + 2 * half + 1) * RS_BN + 16 * t + l];
        acc[t] = __builtin_amdgcn_wmma_f32_16x16x4_f32(
            false, a, false, b, (short)0, acc[t], false, false);
      }
    }
    __syncthreads();
  }

#pragma unroll
  for (int t = 0; t < 8; ++t) {
    const int col = ncol0 + 16 * t + l;
#pragma unroll
    for (int r = 0; r < 8; ++r) {
      const int row = mrow0 + wave * 16 + r + 8 * half;
      C[(size_t)row * ldc + col] = acc[t][r];
    }
  }
}

// f_aug = features + noise / max(||noise_row||, 1e-12)
__global__ void rownorm_aug_kernel(const float* __restrict__ feat,
                                   const float* __restrict__ noise,
                                   float* __restrict__ out) {
  __shared__ float sm[TPB];
  const int i = blockIdx.x, t = threadIdx.x;
  const size_t idx = (size_t)i * DIM + t;
  const float nv = noise[idx];
  const float s  = block_reduce_sum(nv * nv, sm);
  const float nrm = fmaxf(sqrtf(s), 1e-12f);
  out[idx] = feat[idx] + nv / nrm;
}

// s_src[i] = sum_h tanh(xraw[i,h])*att_src[h];  s_dst likewise
__global__ void gat_scores_kernel(const float* __restrict__ xraw,
                                  const float* __restrict__ att_src,
                                  const float* __restrict__ att_dst,
                                  float* s_src, float* s_dst) {
  __shared__ float sm[TPB];
  const int i = blockIdx.x, t = threadIdx.x;
  const float x = tanhf(xraw[(size_t)i * HID + t]);
  const float a = block_reduce_sum(x * att_src[t], sm);
  const float b = block_reduce_sum(x * att_dst[t], sm);
  if (t == 0) { s_src[i] = a; s_dst[i] = b; }
}

// P[i,:] = softmax( mask(adj, leaky_relu(s_src[i]+s_dst[:], 0.05)) )
__global__ void gat_softmax_kernel(const float* __restrict__ adj,
                                   const float* __restrict__ s_src,
                                   const float* __restrict__ s_dst,
                                   float* __restrict__ P) {
  __shared__ float sm[TPB];
  const int i = blockIdx.x, t = threadIdx.x;
  const float si = s_src[i];
  const float* arow = adj + (size_t)i * NND;
  float mx = -3.4e38f;
  for (int j = t; j < NND; j += TPB) {
    float e = si + s_dst[j];
    e = e > 0.f ? e : 0.05f * e;
    if (arow[j] > 0.f) mx = fmaxf(mx, e);
  }
  sm[t] = mx;
  __syncthreads();
  for (int s = TPB >> 1; s > 0; s >>= 1) {
    if (t < s) sm[t] = fmaxf(sm[t], sm[t + s]);
    __syncthreads();
  }
  mx = sm[0];
  __syncthreads();
  float lsum = 0.f;
  for (int j = t; j < NND; j += TPB) {
    float e = si + s_dst[j];
    e = e > 0.f ? e : 0.05f * e;
    if (arow[j] > 0.f) lsum += expf(e - mx);
  }
  const float inv = 1.f / block_reduce_sum(lsum, sm);
  float* prow = P + (size_t)i * NND;
  for (int j = t; j < NND; j += TPB) {
    float e = si + s_dst[j];
    e = e > 0.f ? e : 0.05f * e;
    prow[j] = (arow[j] > 0.f) ? expf(e - mx) * inv : 0.f;
  }
}

__global__ void add2_kernel(float* dst, const float* __restrict__ a) {
  const size_t i = (size_t)blockIdx.x * blockDim.x + threadIdx.x;
  dst[i] += a[i];
}
__global__ void add3_kernel(float* dst, const float* __restrict__ a,
                            const float* __restrict__ b) {
  const size_t i = (size_t)blockIdx.x * blockDim.x + threadIdx.x;
  dst[i] += a[i] + b[i];
}

// row-L2 normalize [N,HID] in place
__global__ void rownorm_inplace_kernel(float* p) {
  __shared__ float sm[TPB];
  const int i = blockIdx.x, t = threadIdx.x;
  const size_t idx = (size_t)i * HID + t;
  const float v = p[idx];
  const float s = block_reduce_sum(v * v, sm);
  p[idx] = v / fmaxf(sqrtf(s), 1e-30f);
}

// rs[i]=sum_j exp(S[i,j]/tau);  rp[i]=sum_j exp(..)*pos[i,j]
__global__ void pair_rows_kernel(const float* __restrict__ S,
                                 const float* __restrict__ pos,
                                 float* rs, float* rp) {
  __shared__ float sm[TPB];
  const int i = blockIdx.x, t = threadIdx.x;
  const float* srow = S + (size_t)i * NND;
  const float* prow = pos + (size_t)i * NND;
  float ls = 0.f, lp = 0.f;
  for (int j = t; j < NND; j += TPB) {
    const float m = expf(srow[j] * (1.f / TAUF));
    ls += m;
    lp += m * prow[j];
  }
  const float s = block_reduce_sum(ls, sm);
  const float p = block_reduce_sum(lp, sm);
  if (t == 0) { rs[i] = s; rp[i] = p; }
}

// cs[j]=sum_i exp(S[i,j]/tau);  cp[j]=sum_i exp(..)*pos[j,i]
__global__ void pair_cols_kernel(const float* __restrict__ S,
                                 const float* __restrict__ pos,
                                 float* cs, float* cp) {
  const int j = blockIdx.x * TPB + threadIdx.x;
  float ls = 0.f, lp = 0.f;
  for (int i = 0; i < NND; ++i) {
    const float m = expf(S[(size_t)i * NND + j] * (1.f / TAUF));
    ls += m;
    lp += m * pos[(size_t)j * NND + i];
  }
  cs[j] = ls;
  cp[j] = lp;
}

__global__ void pair_fin_kernel(const float* rs, const float* rp,
                                const float* cs, const float* cp,
                                float* out_a, float* out_b) {
  __shared__ float sm[TPB];
  const int t = threadIdx.x;
  float la = 0.f, lb = 0.f;
  for (int i = t; i < NND; i += TPB) {
    la += logf(rp[i] / (rs[i] + 1e-8f));
    lb += logf(cp[i] / (cs[i] + 1e-8f));
  }
  const float a = block_reduce_sum(la, sm) * (1.f / NND);
  const float b = block_reduce_sum(lb, sm) * (1.f / NND);
  if (t == 0) { *out_a = a; *out_b = b; }
}

__global__ void zero_kernel(float* p, int n) {
  const int i = blockIdx.x * blockDim.x + threadIdx.x;
  if (i < n) p[i] = 0.f;
}

// out += (1/N) * sum_i sum_h t[i,h]*wp2[h]
__global__ void dotmean_kernel(const float* __restrict__ tmat,
                               const float* __restrict__ wp2, float* out) {
  __shared__ float sm[TPB];
  const int i = blockIdx.x, t = threadIdx.x;
  const float v = tmat[(size_t)i * HID + t] * wp2[t];
  const float s = block_reduce_sum(v, sm);
  if (t == 0) atomicAdd(out, s * (1.f / NND));
}

__global__ void han_combine_kernel(const float* __restrict__ r0,
                                   const float* __restrict__ r1,
                                   const float* __restrict__ r2,
                                   const float* __restrict__ w,
                                   float* __restrict__ out) {
  const size_t i = (size_t)blockIdx.x * blockDim.x + threadIdx.x;
  const float w0 = w[0], w1 = w[1], w2 = w[2];
  const float mx = fmaxf(w0, fmaxf(w1, w2));
  const float e0 = expf(w0 - mx), e1 = expf(w1 - mx), e2 = expf(w2 - mx);
  const float inv = 1.f / (e0 + e1 + e2);
  out[i] = (e0 * r0[i] + e1 * r1[i] + e2 * r2[i]) * inv;
}

__global__ void losses_kernel(const float* sc, float* out) {
  if (threadIdx.x == 0 && blockIdx.x == 0) {
    const float loss1 = LAMF * (-sc[0]) + (1.f - LAMF) * (-sc[1]);
    const float loss2 = LAMF * (-sc[2]) + (1.f - LAMF) * (-sc[3]);
    const float loss3 = 0.5f * (sc[4] + sc[5]);
    out[0] = 0.5f * loss1 + 0.5f * loss2;
    out[1] = loss3;
  }
}

extern "C" void kernel_launch(void* const* d_in, const int* in_sizes, int n_in,
                              void* d_out, int out_size, void* d_ws,
                              size_t ws_size, hipStream_t stream) {
  (void)in_sizes; (void)n_in; (void)out_size; (void)ws_size;
  const float* adj   = (const float*)d_in[0];
  const float* feat  = (const float*)d_in[1];
  const float* pos   = (const float*)d_in[3];   // pos_outer (d_in[2] "pos" unused)
  const float* noise = (const float*)d_in[4];
  const float* Wgat  = (const float*)d_in[5];
  const float* asrc  = (const float*)d_in[6];
  const float* adst  = (const float*)d_in[7];
  const float* Wlin  = (const float*)d_in[8];
  const float* Wres  = (const float*)d_in[9];
  const float* Whmlp = (const float*)d_in[10];
  const float* Wm1   = (const float*)d_in[11];
  const float* bm1   = (const float*)d_in[12];
  const float* Wm2   = (const float*)d_in[13];
  const float* bm2   = (const float*)d_in[14];
  const float* Wx1   = (const float*)d_in[15];
  const float* bx1   = (const float*)d_in[16];
  const float* Wx2   = (const float*)d_in[17];
  const float* bx2   = (const float*)d_in[18];
  const float* Wp1   = (const float*)d_in[19];
  const float* bp1   = (const float*)d_in[20];
  const float* Wp2   = (const float*)d_in[21];
  float* out = (float*)d_out;

  const size_t NN = (size_t)NND * NND;
  const size_t NH = (size_t)NND * HID;   // D == H == 256
  float* w    = (float*)d_ws;
  float* BIG  = w;            // NN floats: P matrix, then similarity matrices
  float* faug = BIG + NN;
  float* tmp  = faug + NH;
  float* res  = tmp + NH;
  float* x0   = res + NH;
  float* x1   = x0 + NH;
  float* x2   = x1 + NH;
  float* r0   = x2 + NH;
  float* r1   = r0 + NH;
  float* r2   = r1 + NH;
  float* pm0  = r2 + NH;
  float* pm1  = pm0 + NH;
  float* pm2  = pm1 + NH;
  float* px1  = pm2 + NH;
  float* px2  = px1 + NH;
  float* ssrc = px2 + NH;
  float* sdst = ssrc + NND;
  float* rs   = sdst + NND;
  float* rp   = rs + NND;
  float* cs   = rp + NND;
  float* cp   = cs + NND;
  float* sc   = cp + NND;     // 6 scalars (a,b per pair)
  float* wsc  = sc + 8;       // 3 HAN logits

  const dim3 blk(TPB);

  auto gemmNT = [&](const float* A, int lda, const float* W, int ldw, float* C,
                    int ldc, const float* bias, float alpha, int accum, int act,
                    int Nc, int K) {
    dim3 g(Nc / BN, NND / BM);
    gemm_nt_f32<<<g, blk, 0, stream>>>(A, lda, W, ldw, C, ldc, bias, alpha,
                                       accum, act, K);
  };
  auto gemmNN = [&](const float* A, int lda, const float* B, int ldb, float* C,
                    int ldc, int Nc, int K) {
    dim3 g(Nc / BN, NND / BM);
    gemm_nn_f32<<<g, blk, 0, stream>>>(A, lda, B, ldb, C, ldc, K);
  };

  // features_aug
  rownorm_aug_kernel<<<NND, TPB, 0, stream>>>(feat, noise, faug);
  // res = 0.2 * F @ Wres^T ;  x2 = relu(F @ Wlin^T)
  gemmNT(feat, DIM, Wres, DIM, res, HID, nullptr, 0.2f, 0, ACT_NONE, HID, DIM);
  gemmNT(feat, DIM, Wlin, DIM, x2, HID, nullptr, 1.0f, 0, ACT_RELU, HID, DIM);

  // --- gat(features) ---
  gemmNT(feat, DIM, Wgat, DIM, tmp, HID, nullptr, 1.0f, 0, ACT_NONE, HID, DIM);
  gat_scores_kernel<<<NND, TPB, 0, stream>>>(tmp, asrc, adst, ssrc, sdst);
  gat_softmax_kernel<<<NND, TPB, 0, stream>>>(adj, ssrc, sdst, BIG);
  gemmNN(BIG, NND, feat, DIM, x0, HID, HID, NND);
  add2_kernel<<<NH / TPB, TPB, 0, stream>>>(x0, res);

  // --- gat(f_aug) ---
  gemmNT(faug, DIM, Wgat, DIM, tmp, HID, nullptr, 1.0f, 0, ACT_NONE, HID, DIM);
  gat_scores_kernel<<<NND, TPB, 0, stream>>>(tmp, asrc, adst, ssrc, sdst);
  gat_softmax_kernel<<<NND, TPB, 0, stream>>>(adj, ssrc, sdst, BIG);
  gemmNN(BIG, NND, faug, DIM, x1, HID, HID, NND);
  gemmNT(faug, DIM, Wres, DIM, tmp, HID, nullptr, 0.2f, 0, ACT_NONE, HID, DIM);
  add3_kernel<<<NH / TPB, TPB, 0, stream>>>(x1, tmp, res);

  // r_k = elu( [left | right] @ Whmlp^T ), Whmlp is [H, 2D] (row stride 2D)
  auto hmlp = [&](const float* left, const float* right, float* r) {
    gemmNT(left, DIM, Whmlp, 2 * DIM, r, HID, nullptr, 1.f, 0, ACT_NONE, HID, DIM);
    gemmNT(right, HID, Whmlp + DIM, 2 * DIM, r, HID, nullptr, 1.f, 1, ACT_ELU, HID, DIM);
  };
  hmlp(feat, x0, r0);
  hmlp(faug, x1, r1);
  hmlp(feat, x2, r2);

  // proj(z) = elu(z@W1^T + b1) @ W2^T + b2
  auto proj = [&](const float* z, const float* W1, const float* b1,
                  const float* W2, const float* b2, float* p) {
    gemmNT(z, HID, W1, HID, tmp, HID, b1, 1.f, 0, ACT_ELU, HID, HID);
    gemmNT(tmp, HID, W2, HID, p, HID, b2, 1.f, 0, ACT_NONE, HID, HID);
  };
  proj(r0, Wm1, bm1, Wm2, bm2, pm0);
  proj(r1, Wm1, bm1, Wm2, bm2, pm1);
  proj(r2, Wm1, bm1, Wm2, bm2, pm2);
  proj(r1, Wx1, bx1, Wx2, bx2, px1);
  proj(r2, Wx1, bx1, Wx2, bx2, px2);

  rownorm_inplace_kernel<<<NND, TPB, 0, stream>>>(pm0);
  rownorm_inplace_kernel<<<NND, TPB, 0, stream>>>(pm1);
  rownorm_inplace_kernel<<<NND, TPB, 0, stream>>>(pm2);
  rownorm_inplace_kernel<<<NND, TPB, 0, stream>>>(px1);
  rownorm_inplace_kernel<<<NND, TPB, 0, stream>>>(px2);

  // pair losses: S = p̂a @ p̂b^T; reduce exp(S/tau) by rows/cols with pos
  auto pair = [&](const float* pa, const float* pb, float* oa, float* ob) {
    gemmNT(pa, HID, pb, HID, BIG, NND, nullptr, 1.f, 0, ACT_NONE, NND, HID);
    pair_rows_kernel<<<NND, TPB, 0, stream>>>(BIG, pos, rs, rp);
    pair_cols_kernel<<<NND / TPB, TPB, 0, stream>>>(BIG, pos, cs, cp);
    pair_fin_kernel<<<1, TPB, 0, stream>>>(rs, rp, cs, cp, oa, ob);
  };
  pair(pm0, pm1, sc + 0, sc + 1);
  pair(pm0, pm2, sc + 2, sc + 3);
  pair(px1, px2, sc + 4, sc + 5);

  // HAN semantic attention
  zero_kernel<<<1, 32, 0, stream>>>(wsc, 3);
  const float* rlist[3] = {r0, r1, r2};
  for (int k = 0; k < 3; ++k) {
    gemmNT(rlist[k], HID, Wp1, HID, tmp, HID, bp1, 1.f, 0, ACT_TANH, HID, HID);
    dotmean_kernel<<<NND, TPB, 0, stream>>>(tmp, Wp2, wsc + k);
  }
  han_combine_kernel<<<NH / TPB, TPB, 0, stream>>>(r0, r1, r2, wsc, out);
  losses_kernel<<<1, 32, 0, stream>>>(sc, out + NH);
}